// HGNLayer_18734647345348
// MI455X (gfx1250) — compile-verified
//
#include <hip/hip_runtime.h>
#include <math.h>

#define E_DIM 2048
#define R_DIM 24
#define F_DIM 128
#define NEG_SLOPE 0.2f
#define MT_STRIDE 132   // 128 + 4 padding to dodge LDS bank conflicts

typedef __attribute__((ext_vector_type(2))) float v2f;
typedef __attribute__((ext_vector_type(8))) float v8f;

// ---- CDNA5 f32 WMMA: D(16x16 f32) = A(16x4) * B(4x16) + C ----
// A/B per-lane layout: m/n = lane&15 ; VGPR0 holds K = (lane>>4)*2, VGPR1 holds K+1.
__device__ __forceinline__ v8f wmma4(v2f a, v2f b, v8f c) {
  return __builtin_amdgcn_wmma_f32_16x16x4_f32(false, a, false, b, (short)0, c,
                                               false, false);
}

__device__ __forceinline__ float att_val(float s, float e, float r) {
  float x = s * e * r;
  return x > 0.0f ? x : NEG_SLOPE * x;
}

// ---------------- K0: adjS[e,:] = softmax(A[e,:]) over R ----------------
__global__ void k_adj_softmax(const float* __restrict__ A, float* __restrict__ adjS) {
  int e = blockIdx.x * blockDim.x + threadIdx.x;
  if (e >= E_DIM) return;
  const float* row = A + e * R_DIM;
  float m = -INFINITY;
  for (int r = 0; r < R_DIM; ++r) m = fmaxf(m, row[r]);
  float s = 0.f;
  for (int r = 0; r < R_DIM; ++r) s += __expf(row[r] - m);
  float inv = 1.0f / s;
  float* o = adjS + e * R_DIM;
  for (int r = 0; r < R_DIM; ++r) o[r] = __expf(row[r] - m) * inv;
}

// ---------------- small utility: zero n floats at p ----------------
__global__ void k_zero(float* __restrict__ p, int n) {
  int i = blockIdx.x * blockDim.x + threadIdx.x;
  if (i < n) p[i] = 0.f;
}

// ---------------- K1: per-(r,f) max & sumexp over e (att_E stats) ----------------
__global__ void __launch_bounds__(256) k_col_stats(const float* __restrict__ ent,
                                                   const float* __restrict__ rel,
                                                   const float* __restrict__ adjS,
                                                   float* __restrict__ colMax,
                                                   float* __restrict__ colSum) {
  __shared__ float sm[256];
  int cid = blockIdx.x;             // cid = r*F + f
  int r = cid / F_DIM, f = cid % F_DIM;
  float relv = rel[cid];
  int t = threadIdx.x;
  float m = -INFINITY;
  for (int e = t; e < E_DIM; e += 256)
    m = fmaxf(m, att_val(adjS[e * R_DIM + r], ent[e * F_DIM + f], relv));
  sm[t] = m; __syncthreads();
  for (int s = 128; s; s >>= 1) { if (t < s) sm[t] = fmaxf(sm[t], sm[t + s]); __syncthreads(); }
  m = sm[0]; __syncthreads();
  float acc = 0.f;
  for (int e = t; e < E_DIM; e += 256)
    acc += __expf(att_val(adjS[e * R_DIM + r], ent[e * F_DIM + f], relv) - m);
  sm[t] = acc; __syncthreads();
  for (int s = 128; s; s >>= 1) { if (t < s) sm[t] += sm[t + s]; __syncthreads(); }
  if (t == 0) { colMax[cid] = m; colSum[cid] = sm[0]; }
}

// ---------------- K2: per-(e,f) max & sumexp over r (att_R stats) ----------------
__global__ void __launch_bounds__(256) k_row_stats(const float* __restrict__ ent,
                                                   const float* __restrict__ rel,
                                                   const float* __restrict__ adjS,
                                                   float* __restrict__ rowM,
                                                   float* __restrict__ rowS) {
  int idx = blockIdx.x * blockDim.x + threadIdx.x;
  if (idx >= E_DIM * F_DIM) return;
  int e = idx / F_DIM, f = idx % F_DIM;
  float ef = ent[idx];
  float m = -INFINITY;
  for (int r = 0; r < R_DIM; ++r)
    m = fmaxf(m, att_val(adjS[e * R_DIM + r], ef, rel[r * F_DIM + f]));
  float s = 0.f;
  for (int r = 0; r < R_DIM; ++r)
    s += __expf(att_val(adjS[e * R_DIM + r], ef, rel[r * F_DIM + f]) - m);
  rowM[idx] = m; rowS[idx] = s;
}

// ---------------- K3: h_prime[e,f] = ent[e,f] * sum_r att_E[e,r,f] ----------------
__global__ void __launch_bounds__(256) k_hprime(const float* __restrict__ ent,
                                                const float* __restrict__ rel,
                                                const float* __restrict__ adjS,
                                                const float* __restrict__ colMax,
                                                const float* __restrict__ colSum,
                                                float* __restrict__ hprime) {
  int idx = blockIdx.x * blockDim.x + threadIdx.x;
  if (idx >= E_DIM * F_DIM) return;
  int e = idx / F_DIM, f = idx % F_DIM;
  float ef = ent[idx];
  float h = 0.f;
  for (int r = 0; r < R_DIM; ++r) {
    int cid = r * F_DIM + f;
    h += __expf(att_val(adjS[e * R_DIM + r], ef, rel[cid]) - colMax[cid]) / colSum[cid];
  }
  hprime[idx] = h * ef;
}

// ---------------- K4: r_prime[r,f] = rel[r,f] * sum_e att_R[e,r,f] ----------------
__global__ void __launch_bounds__(256) k_rprime(const float* __restrict__ ent,
                                                const float* __restrict__ rel,
                                                const float* __restrict__ adjS,
                                                const float* __restrict__ rowM,
                                                const float* __restrict__ rowS,
                                                float* __restrict__ rprime) {
  __shared__ float sm[256];
  int cid = blockIdx.x;             // r*F + f
  int r = cid / F_DIM, f = cid % F_DIM;
  float relv = rel[cid];
  int t = threadIdx.x;
  float acc = 0.f;
  for (int e = t; e < E_DIM; e += 256) {
    int ef = e * F_DIM + f;
    acc += __expf(att_val(adjS[e * R_DIM + r], ent[ef], relv) - rowM[ef]) / rowS[ef];
  }
  sm[t] = acc; __syncthreads();
  for (int s = 128; s; s >>= 1) { if (t < s) sm[t] += sm[t + s]; __syncthreads(); }
  if (t == 0) rprime[cid] = sm[0] * relv;
}

// ---------------- K5: alphaPre[e*R+r] = <att_R[e,r,:], lin_w> + b (wave per (e,r)) ----------------
__global__ void k_alpha_pre(const float* __restrict__ ent, const float* __restrict__ rel,
                            const float* __restrict__ adjS,
                            const float* __restrict__ rowM, const float* __restrict__ rowS,
                            const float* __restrict__ lin_w, const float* __restrict__ lin_b,
                            float* __restrict__ alphaPre) {
  int lane = threadIdx.x;                                   // 0..31 (wave32)
  int pair = blockIdx.x * blockDim.y + threadIdx.y;         // (e,r) pair, e-major
  if (pair >= E_DIM * R_DIM) return;
  int e = pair / R_DIM, r = pair % R_DIM;
  float a_er = adjS[pair];
  float s = 0.f;
  #pragma unroll
  for (int q = 0; q < 4; ++q) {
    int f = lane + q * 32;
    int ef = e * F_DIM + f;
    float v = __expf(att_val(a_er, ent[ef], rel[r * F_DIM + f]) - rowM[ef]) / rowS[ef];
    s += v * lin_w[f];
  }
  for (int off = 16; off; off >>= 1) s += __shfl_xor(s, off, 32);
  if (lane == 0) alphaPre[pair] = s + lin_b[0];
}

// ---------------- K6: raw reshape (E,R)->(R,E), softmax over E = softmax over
// contiguous groups of E elements of the flat e-major array ----------------
__global__ void __launch_bounds__(256) k_alpha_softmax(const float* __restrict__ alphaPre,
                                                       float* __restrict__ alphaS) {
  __shared__ float sm[256];
  int g = blockIdx.x;                       // group r' : flat [g*E, (g+1)*E)
  const float* src = alphaPre + (size_t)g * E_DIM;
  float* dst = alphaS + (size_t)g * E_DIM;
  int t = threadIdx.x;
  float m = -INFINITY;
  for (int i = t; i < E_DIM; i += 256) m = fmaxf(m, src[i]);
  sm[t] = m; __syncthreads();
  for (int s = 128; s; s >>= 1) { if (t < s) sm[t] = fmaxf(sm[t], sm[t + s]); __syncthreads(); }
  m = sm[0]; __syncthreads();
  float acc = 0.f;
  for (int i = t; i < E_DIM; i += 256) acc += __expf(src[i] - m);
  sm[t] = acc; __syncthreads();
  for (int s = 128; s; s >>= 1) { if (t < s) sm[t] += sm[t + s]; __syncthreads(); }
  float inv = 1.0f / sm[0];
  for (int i = t; i < E_DIM; i += 256) dst[i] = __expf(src[i] - m) * inv;
}

// ---------------- generic WMMA f32 GEMM: Out[M x 128] = A[M x K] * B[K x 128] ----------------
// A must be readable (zero-padded) up to gridDim.x*16 rows; Mvalid only guards stores.
__global__ void k_gemm_f32(const float* __restrict__ Amat, const float* __restrict__ Bmat,
                           float* __restrict__ Out, int Mvalid, int Kdim) {
  int lane = threadIdx.x;
  int m0 = blockIdx.x * 16;
  int n0 = (blockIdx.y * blockDim.y + threadIdx.y) * 16;
  int colN = n0 + (lane & 15);
  int rowSel = lane >> 4;
  int mRow = m0 + (lane & 15);
  const float* aRow = Amat + (size_t)mRow * Kdim + rowSel * 2;
  v8f c = {};
  #pragma unroll 4
  for (int k = 0; k < Kdim; k += 4) {
    int kk = k + rowSel * 2;
    v2f a = *(const v2f*)(aRow + k);            // global_load_b64
    v2f b;
    b.x = Bmat[kk * F_DIM + colN];
    b.y = Bmat[(kk + 1) * F_DIM + colN];
    c = wmma4(a, b, c);
  }
  int rowBase = rowSel * 8;
  #pragma unroll
  for (int v = 0; v < 8; ++v) {
    int row = m0 + rowBase + v;
    if (row < Mvalid) Out[row * F_DIM + colN] = c[v];
  }
}

// ---------------- K8: fused streaming kernel (the bandwidth-bound one) ----------------
// Per block: 16 output rows. Single exactly-once pass over 403MB adjacency:
//   Mtile[i][j] = sum_r alpha[r,i]*adj[r,i,j]  (LDS, chunked over j)
//   C += Mtile @ EW[chunk,:]  via v_wmma_f32_16x16x4_f32, C initialized with h_prime.
__global__ void __launch_bounds__(256) k_ent_fused(const float* __restrict__ adjc,
                                                   const float* __restrict__ alphaS,
                                                   const float* __restrict__ EW,
                                                   const float* __restrict__ hprime,
                                                   float* __restrict__ outEnt) {
  __shared__ float Mtile[16 * MT_STRIDE];
  __shared__ float alphaT[R_DIM * 16];
  const int lane = threadIdx.x;             // 0..31
  const int w = threadIdx.y;                // 0..7 (wave id == N-tile)
  const int tid = w * 32 + lane;            // 0..255
  const int i0 = blockIdx.x * 16;

  // Stage alpha[r, i0..i0+15] into LDS with the gfx1250 async global->LDS path.
  // Low 32 bits of a generic LDS pointer are the LDS byte address (aperture rule).
  for (int idx = tid; idx < R_DIM * 16; idx += 256) {
    int r = idx >> 4, ii = idx & 15;
    unsigned ldsAddr = (unsigned)(unsigned long long)(alphaT + idx);
    const float* gp = alphaS + (size_t)r * E_DIM + i0 + ii;
    asm volatile("global_load_async_to_lds_b32 %0, %1, off"
                 :: "v"(ldsAddr), "v"(gp) : "memory");
  }
  asm volatile("s_wait_asynccnt 0x0" ::: "memory");
  __syncthreads();

  const int colN = w * 16 + (lane & 15);
  const int rowSel = lane >> 4;
  const int rowBase = rowSel * 8;
  v8f c;
  #pragma unroll
  for (int v = 0; v < 8; ++v) c[v] = hprime[(i0 + rowBase + v) * F_DIM + colN];

  const int iRow = tid >> 4;                // 0..15
  const int jStart = (tid & 15) * 8;        // 0..120, float4-pair per thread

  for (int kc = 0; kc < E_DIM; kc += 128) {
    float acc[8];
    #pragma unroll
    for (int q = 0; q < 8; ++q) acc[q] = 0.f;
    #pragma unroll 4
    for (int r = 0; r < R_DIM; ++r) {
      const float* base = adjc + ((size_t)(r * E_DIM + i0 + iRow)) * E_DIM + kc + jStart;
      float al = alphaT[r * 16 + iRow];
      const float4* p = (const float4*)base;
      float4 x0 = p[0];
      float4 x1 = p[1];
      if (kc + 128 < E_DIM) __builtin_prefetch(base + 128, 0, 0);  // next chunk -> global_prefetch_b8
      acc[0] += al * x0.x; acc[1] += al * x0.y; acc[2] += al * x0.z; acc[3] += al * x0.w;
      acc[4] += al * x1.x; acc[5] += al * x1.y; acc[6] += al * x1.z; acc[7] += al * x1.w;
    }
    #pragma unroll
    for (int q = 0; q < 8; ++q) Mtile[iRow * MT_STRIDE + jStart + q] = acc[q];
    __syncthreads();
    #pragma unroll 8
    for (int k4 = 0; k4 < 128; k4 += 4) {
      int kk = k4 + rowSel * 2;
      v2f a = *(const v2f*)&Mtile[(lane & 15) * MT_STRIDE + kk];  // ds_load_b64
      v2f b;
      b.x = EW[(kc + kk) * F_DIM + colN];
      b.y = EW[(kc + kk + 1) * F_DIM + colN];
      c = wmma4(a, b, c);
    }
    __syncthreads();
  }
  #pragma unroll
  for (int v = 0; v < 8; ++v) outEnt[(i0 + rowBase + v) * F_DIM + colN] = c[v];
}

extern "C" void kernel_launch(void* const* d_in, const int* in_sizes, int n_in,
                              void* d_out, int out_size, void* d_ws, size_t ws_size,
                              hipStream_t stream) {
  (void)in_sizes; (void)n_in; (void)out_size; (void)ws_size;
  const float* ent  = (const float*)d_in[0];
  const float* rel  = (const float*)d_in[1];
  const float* adjc = (const float*)d_in[2];
  const float* A    = (const float*)d_in[3];
  const float* Went = (const float*)d_in[4];
  const float* Wrel = (const float*)d_in[5];
  const float* lw   = (const float*)d_in[6];
  const float* lb   = (const float*)d_in[7];

  float* out    = (float*)d_out;
  float* outEnt = out;
  float* outRel = out + (size_t)E_DIM * F_DIM;

  float* ws = (float*)d_ws;
  float* adjS     = ws; ws += E_DIM * R_DIM;
  float* rowM     = ws; ws += E_DIM * F_DIM;
  float* rowS     = ws; ws += E_DIM * F_DIM;
  float* colMax   = ws; ws += R_DIM * F_DIM;
  float* colSum   = ws; ws += R_DIM * F_DIM;
  float* hprime   = ws; ws += E_DIM * F_DIM;
  float* rprime   = ws; ws += 32 * F_DIM;      // padded to 32 rows for guard-free GEMM
  float* alphaPre = ws; ws += E_DIM * R_DIM;
  float* alphaS   = ws; ws += E_DIM * R_DIM;
  float* EW       = ws; ws += E_DIM * F_DIM;

  k_adj_softmax<<<E_DIM / 256, 256, 0, stream>>>(A, adjS);
  k_zero<<<4, 256, 0, stream>>>(rprime + R_DIM * F_DIM, (32 - R_DIM) * F_DIM);
  k_col_stats<<<R_DIM * F_DIM, 256, 0, stream>>>(ent, rel, adjS, colMax, colSum);
  k_row_stats<<<(E_DIM * F_DIM) / 256, 256, 0, stream>>>(ent, rel, adjS, rowM, rowS);
  k_hprime<<<(E_DIM * F_DIM) / 256, 256, 0, stream>>>(ent, rel, adjS, colMax, colSum, hprime);
  k_rprime<<<R_DIM * F_DIM, 256, 0, stream>>>(ent, rel, adjS, rowM, rowS, rprime);
  k_alpha_pre<<<(E_DIM * R_DIM) / 8, dim3(32, 8), 0, stream>>>(ent, rel, adjS, rowM, rowS,
                                                               lw, lb, alphaPre);
  k_alpha_softmax<<<R_DIM, 256, 0, stream>>>(alphaPre, alphaS);
  // EW = ent @ weight_ent  (so the fused kernel does M @ EW == (M @ ent) @ Went)
  k_gemm_f32<<<dim3(E_DIM / 16, 2), dim3(32, 4), 0, stream>>>(ent, Went, EW, E_DIM, F_DIM);
  // fused: single pass over adjacency, WMMA GEMM, h_prime epilogue
  k_ent_fused<<<E_DIM / 16, dim3(32, 8), 0, stream>>>(adjc, alphaS, EW, hprime, outEnt);
  // rel_output = r_prime @ weight_rel (rows 24..31 zero-padded, store-guarded)
  k_gemm_f32<<<dim3(2, 2), dim3(32, 4), 0, stream>>>(rprime, Wrel, outRel, R_DIM, F_DIM);
}